// VectorQuantizer_42477226557441
// MI455X (gfx1250) — compile-verified
//
#include <hip/hip_runtime.h>
#include <math.h>

#define N_E   2048
#define EDIM  128
#define NTOK  8192
#define NELEM (N_E * NTOK)          // 16,777,216 distance-matrix elements
#define OUTQ  (NTOK * EDIM)         // 1,048,576 x_q elements

typedef __attribute__((ext_vector_type(2))) float v2f;
typedef __attribute__((ext_vector_type(8))) float v8f;

// ---------------------------------------------------------------- init
__global__ void k_init(unsigned* minmax, float* acc, int* bad) {
    minmax[0] = 0u;           // running max (positive-float uint ordering)
    minmax[1] = 0xFFFFFFFFu;  // running min
    *acc = 0.0f;
    *bad = 0;
}

__global__ void k_ones(double* v) {
    int j = blockIdx.x * 256 + threadIdx.x;
    if (j < NTOK) v[j] = 1.0;
}

// ------------------------------------------------- squared row norms
// blocks [0,NTOK): rows of x ; blocks [NTOK,NTOK+N_E): rows of codebook
__global__ void k_sqnorm(const float* __restrict__ x, const float* __restrict__ cb,
                         float* __restrict__ nx2, float* __restrict__ ne2) {
    __shared__ float s[128];
    int b = blockIdx.x, t = threadIdx.x;
    const float* row;
    float* out;
    if (b < NTOK) { row = x + (size_t)b * EDIM;          out = nx2 + b; }
    else          { row = cb + (size_t)(b - NTOK) * EDIM; out = ne2 + (b - NTOK); }
    float val = row[t];
    s[t] = val * val;
    __syncthreads();
    for (int off = 64; off > 0; off >>= 1) {
        if (t < off) s[t] += s[t + off];
        __syncthreads();
    }
    if (t == 0) *out = s[0];
}

// ------------------------------------------------- WMMA distance GEMM
// One 16x16 (codes x tokens) tile per wave32; K=128 via 32x V_WMMA_F32_16X16X4_F32.
// d_t[i][j] = ||x_j||^2 + ||e_i||^2 - 2 e_i.x_j   (i-major for the Sinkhorn passes)
__global__ void k_gemm(const float* __restrict__ x, const float* __restrict__ cb,
                       const float* __restrict__ nx2, const float* __restrict__ ne2,
                       float* __restrict__ dt, unsigned* __restrict__ minmax) {
    int wave = threadIdx.x >> 5;
    int lane = threadIdx.x & 31;
    int gtile = blockIdx.x * 8 + wave;          // 65536 tiles total
    int i0 = (gtile & 127) * 16;                // code tile origin
    int j0 = (gtile >> 7) * 16;                 // token tile origin

    int m    = lane & 15;
    int koff = (lane >> 4) * 2;                 // A/B: lanes 0-15 hold K+0..1, 16-31 hold K+2..3
    const float* arow = cb + (size_t)(i0 + m) * EDIM;   // A rows = codebook
    const float* brow = x  + (size_t)(j0 + m) * EDIM;   // B cols = tokens

    v8f c = {};
    #pragma unroll
    for (int k = 0; k < EDIM; k += 4) {
        v2f a, b;
        a.x = arow[k + koff]; a.y = arow[k + koff + 1];
        b.x = brow[k + koff]; b.y = brow[k + koff + 1];
        c = __builtin_amdgcn_wmma_f32_16x16x4_f32(false, a, false, b,
                                                  (short)0, c, false, false);
    }

    // C/D layout: VGPR r, lanes 0-15 -> M=r ; lanes 16-31 -> M=r+8 ; N = lane&15
    float nx   = nx2[j0 + (lane & 15)];
    float lmax = -1e30f, lmin = 1e30f;
    #pragma unroll
    for (int r = 0; r < 8; r++) {
        int M = r + ((lane >> 4) << 3);
        int N = lane & 15;
        float dval = nx + ne2[i0 + M] - 2.0f * c[r];
        dt[(size_t)(i0 + M) * NTOK + (j0 + N)] = dval;
        lmax = fmaxf(lmax, dval);
        lmin = fminf(lmin, dval);
    }

    __shared__ float smax[256], smin[256];
    smax[threadIdx.x] = lmax; smin[threadIdx.x] = lmin;
    __syncthreads();
    for (int off = 128; off > 0; off >>= 1) {
        if (threadIdx.x < off) {
            smax[threadIdx.x] = fmaxf(smax[threadIdx.x], smax[threadIdx.x + off]);
            smin[threadIdx.x] = fminf(smin[threadIdx.x], smin[threadIdx.x + off]);
        }
        __syncthreads();
    }
    if (threadIdx.x == 0) {
        atomicMax(&minmax[0], __float_as_uint(smax[0]));  // d > 0 always
        atomicMin(&minmax[1], __float_as_uint(smin[0]));
    }
}

// --------------------------------------- E = exp(-(d-middle)/(amp*eps))  [f64]
__global__ void k_expgen(const float* __restrict__ dt, double* __restrict__ E,
                         const unsigned* __restrict__ minmax) {
    float mx = __uint_as_float(minmax[0]);
    float mn = __uint_as_float(minmax[1]);
    float middle = (mx + mn) * 0.5f;
    float amp    = mx - middle + 1e-5f;
    const double coef = -1.0 / 0.003;           // -1/eps
    size_t base = (size_t)blockIdx.x * 2048 + threadIdx.x;
    #pragma unroll
    for (int it = 0; it < 8; it++) {
        size_t idx = base + (size_t)it * 256;
        double l = (double)((dt[idx] - middle) / amp) * coef;   // |l| <= ~333: f64 only
        E[idx] = exp(l);
    }
}

// ------------------------------------------- Sinkhorn row pass: u = 1/(K * E v)
__global__ void k_row(const double* __restrict__ E, const double* __restrict__ v,
                      double* __restrict__ u) {
    const double* row = E + (size_t)blockIdx.x * NTOK;
    double acc = 0.0;
    for (int j = threadIdx.x; j < NTOK; j += 256) acc += row[j] * v[j];
    __shared__ double s[256];
    s[threadIdx.x] = acc;
    __syncthreads();
    for (int off = 128; off > 0; off >>= 1) {
        if (threadIdx.x < off) s[threadIdx.x] += s[threadIdx.x + off];
        __syncthreads();
    }
    if (threadIdx.x == 0) u[blockIdx.x] = 1.0 / ((double)N_E * s[0]);
}

// -------------------------------- Sinkhorn col pass (split over 8 i-segments)
__global__ void k_colpart(const double* __restrict__ E, const double* __restrict__ u,
                          double* __restrict__ vpart) {
    __shared__ double su[256];
    int ibase = blockIdx.y * 256;
    su[threadIdx.x] = u[ibase + threadIdx.x];
    __syncthreads();
    int j = blockIdx.x * 256 + threadIdx.x;
    double acc = 0.0;
    #pragma unroll 4
    for (int i = 0; i < 256; i++)
        acc += E[(size_t)(ibase + i) * NTOK + j] * su[i];
    vpart[(size_t)blockIdx.y * NTOK + j] = acc;
}

__global__ void k_colfin(const double* __restrict__ vpart, double* __restrict__ v) {
    int j = blockIdx.x * 256 + threadIdx.x;
    double acc = 0.0;
    #pragma unroll
    for (int s = 0; s < 8; s++) acc += vpart[(size_t)s * NTOK + j];
    v[j] = 1.0 / ((double)NTOK * acc);
}

// ------------------------------------------------- NaN/Inf fallback detection
__global__ void k_bad(const double* __restrict__ u, const double* __restrict__ v,
                      int* __restrict__ bad) {
    int local = 0;
    for (int i = threadIdx.x; i < N_E;  i += 256) if (!isfinite(u[i])) local = 1;
    for (int j = threadIdx.x; j < NTOK; j += 256) if (!isfinite(v[j])) local = 1;
    if (local) atomicOr(bad, 1);
}

// ------------------- indices: argmax_i (log u_i + log E_ij); fallback argmax log E
// (column scaling v_j never changes the per-token argmax)
__global__ void k_argmax(const double* __restrict__ E, const double* __restrict__ u,
                         const int* __restrict__ bad, int* __restrict__ idxout) {
    __shared__ double slogu[N_E];               // 16 KB LDS
    for (int i = threadIdx.x; i < N_E; i += 256) slogu[i] = log(u[i]);
    __syncthreads();
    int j = blockIdx.x * 256 + threadIdx.x;
    int useBad = *bad;
    double best = -1.0e308;
    int bi = 0;
    for (int i = 0; i < N_E; i++) {
        double le = log(E[(size_t)i * NTOK + j]);   // == -d_c/eps, never 0/Inf
        double score = useBad ? le : (slogu[i] + le);
        if (score > best) { best = score; bi = i; }
    }
    idxout[j] = bi;
}

// ---------------------------- outputs: x_q gather, loss partials, index floats
__global__ void k_out(const float* __restrict__ x, const float* __restrict__ cb,
                      const int* __restrict__ idx, float* __restrict__ out,
                      float* __restrict__ acc) {
    int gid = blockIdx.x * 256 + threadIdx.x;   // over NTOK*EDIM
    int j = gid >> 7, k = gid & 127;
    int ci = idx[j];
    float q = cb[(size_t)ci * EDIM + k];
    out[gid] = q;                               // straight-through == x_q numerically
    float diff = q - x[gid];
    __shared__ float s[256];
    s[threadIdx.x] = diff * diff;
    __syncthreads();
    for (int off = 128; off > 0; off >>= 1) {
        if (threadIdx.x < off) s[threadIdx.x] += s[threadIdx.x + off];
        __syncthreads();
    }
    if (threadIdx.x == 0) atomicAdd(acc, s[0]);
    if (k == 0) out[OUTQ + 1 + j] = (float)ci;  // indices, flattened after loss
}

__global__ void k_loss(const float* __restrict__ acc, float* __restrict__ out) {
    // loss = codebook_loss + 0.25*commitment = 1.25 * mse
    out[OUTQ] = 1.25f * (*acc) / (float)OUTQ;
}

// ===================================================================
extern "C" void kernel_launch(void* const* d_in, const int* in_sizes, int n_in,
                              void* d_out, int out_size, void* d_ws, size_t ws_size,
                              hipStream_t stream) {
    (void)in_sizes; (void)n_in; (void)out_size; (void)ws_size;
    const float* x  = (const float*)d_in[0];   // [8192,128]
    const float* cb = (const float*)d_in[1];   // [2048,128]
    float* out = (float*)d_out;

    char* ws = (char*)d_ws;
    float*    dt     = (float*)   (ws);                      // 64 MB  d[i][j]
    double*   E      = (double*)  (ws + 67108864);           // 128 MB exp matrix (L2-resident)
    double*   u      = (double*)  (ws + 201326592);          // 16 KB
    double*   v      = (double*)  (ws + 201342976);          // 64 KB
    double*   vpart  = (double*)  (ws + 201408512);          // 512 KB
    float*    nx2    = (float*)   (ws + 201932800);          // 32 KB
    float*    ne2    = (float*)   (ws + 201965568);          // 8 KB
    int*      idx    = (int*)     (ws + 201973760);          // 32 KB
    unsigned* minmax = (unsigned*)(ws + 202006528);
    float*    acc    = (float*)   (ws + 202006536);
    int*      bad    = (int*)     (ws + 202006540);

    k_init  <<<1, 1, 0, stream>>>(minmax, acc, bad);
    k_ones  <<<NTOK / 256, 256, 0, stream>>>(v);
    k_sqnorm<<<NTOK + N_E, 128, 0, stream>>>(x, cb, nx2, ne2);
    k_gemm  <<<(N_E / 16) * (NTOK / 16) / 8, 256, 0, stream>>>(x, cb, nx2, ne2, dt, minmax);
    k_expgen<<<NELEM / 2048, 256, 0, stream>>>(dt, E, minmax);

    for (int it = 0; it < 100; it++) {
        k_row    <<<N_E, 256, 0, stream>>>(E, v, u);
        k_colpart<<<dim3(NTOK / 256, 8), 256, 0, stream>>>(E, u, vpart);
        k_colfin <<<NTOK / 256, 256, 0, stream>>>(vpart, v);
    }

    k_bad   <<<1, 256, 0, stream>>>(u, v, bad);
    k_argmax<<<NTOK / 256, 256, 0, stream>>>(E, u, bad, idx);
    k_out   <<<OUTQ / 256, 256, 0, stream>>>(x, cb, idx, out, acc);
    k_loss  <<<1, 1, 0, stream>>>(acc, out);
}